// GraphAttentionLayer_38886633898682
// MI455X (gfx1250) — compile-verified
//
#include <hip/hip_runtime.h>
#include <math.h>
#include <stdint.h>

typedef __attribute__((ext_vector_type(2))) float v2f;
typedef __attribute__((ext_vector_type(4))) float v4f;
typedef __attribute__((ext_vector_type(8))) float v8f;

#define IN_F   128
#define OUT_F  128
#define B_DIM  8
#define T_DIM  2048
#define ALPHA  0.2f
#define TJ     64
#define LDSP   132   // padded LDS row stride (floats) to avoid 64-bank conflicts

__device__ __forceinline__ v8f wmma_f32(v2f a, v2f b, v8f c) {
  // D = A(16x4 f32) * B(4x16 f32) + C(16x16 f32), exact f32 path on CDNA5
  return __builtin_amdgcn_wmma_f32_16x16x4_f32(
      /*neg_a=*/false, a, /*neg_b=*/false, b,
      /*c_mod=*/(short)0, c, /*reuse_a=*/false, /*reuse_b=*/false);
}

// CDNA5 async DMA: global -> LDS, 16B per lane, tracked by ASYNCcnt.
__device__ __forceinline__ void async_ld_b128(uint32_t lds_byte, const float* gsrc) {
  asm volatile("global_load_async_to_lds_b128 %0, %1, off"
               :: "v"(lds_byte), "v"(gsrc) : "memory");
}
__device__ __forceinline__ void wait_async0() {
  asm volatile("s_wait_asynccnt 0x0" ::: "memory");
}

// ---------------------------------------------------------------------------
// Kernel A: h = x @ W   ((B*T) x 128) = ((B*T) x 128) @ (128 x 128)
// 1 workgroup = 8 waves = 128 rows; W staged in LDS.
// ---------------------------------------------------------------------------
__global__ __launch_bounds__(256) void gat_xw(const float* __restrict__ x,
                                              const float* __restrict__ W,
                                              float* __restrict__ h) {
  __shared__ __align__(16) float sW[IN_F * OUT_F];  // 64 KB
  for (int t = threadIdx.x; t < IN_F * OUT_F / 4; t += 256)
    *(v4f*)(sW + t * 4) = *(const v4f*)(W + t * 4);
  __syncthreads();

  const int lane = threadIdx.x & 31;
  const int wave = threadIdx.x >> 5;
  const int half = lane >> 4;   // 0 or 1
  const int l16  = lane & 15;
  const int row0 = blockIdx.x * 128 + wave * 16;
  const float* xrow = x + (size_t)(row0 + l16) * IN_F;  // A: lane -> row M = l16

  v8f acc[8] = {};  // 8 N-tiles of 16 -> full 16x128 output tile

  for (int k0 = 0; k0 < IN_F; k0 += 4) {
    // A fragment 16x4: VGPR0 = K = k0+2*half, VGPR1 = K = k0+2*half+1
    v2f a;
    a.x = xrow[k0 + 2 * half + 0];
    a.y = xrow[k0 + 2 * half + 1];
    const float* w0 = sW + (size_t)(k0 + 2 * half) * OUT_F;
    #pragma unroll
    for (int nt = 0; nt < 8; ++nt) {
      v2f b;  // B fragment 4x16: lane -> N = l16, half selects K pair
      b.x = w0[nt * 16 + l16];
      b.y = w0[OUT_F + nt * 16 + l16];
      acc[nt] = wmma_f32(a, b, acc[nt]);
    }
  }
  // C layout: VGPR r, lane -> M = r + 8*half, N = l16
  #pragma unroll
  for (int r = 0; r < 8; ++r) {
    float* orow = h + (size_t)(row0 + r + 8 * half) * OUT_F;
    #pragma unroll
    for (int nt = 0; nt < 8; ++nt)
      orow[nt * 16 + l16] = acc[nt][r];
  }
}

// ---------------------------------------------------------------------------
// Kernel B: f1 = h@a1, f2 = h@a2, fmax[b] = max_j f2[b,j]
// One workgroup per batch.
// ---------------------------------------------------------------------------
__global__ __launch_bounds__(256) void gat_fvec(const float* __restrict__ h,
                                                const float* __restrict__ a,
                                                float* __restrict__ f1,
                                                float* __restrict__ f2,
                                                float* __restrict__ fmax) {
  __shared__ float sa1[OUT_F], sa2[OUT_F], sred[256];
  if (threadIdx.x < OUT_F) {
    sa1[threadIdx.x] = a[threadIdx.x];
    sa2[threadIdx.x] = a[OUT_F + threadIdx.x];
  }
  __syncthreads();

  const int b = blockIdx.x;
  float lmax = -3.4e38f;
  for (int r = threadIdx.x; r < T_DIM; r += 256) {
    const float* hr = h + ((size_t)b * T_DIM + r) * OUT_F;
    float s1 = 0.f, s2 = 0.f;
    #pragma unroll 4
    for (int n = 0; n < OUT_F; ++n) {
      float v = hr[n];
      s1 = fmaf(v, sa1[n], s1);
      s2 = fmaf(v, sa2[n], s2);
    }
    f1[b * T_DIM + r] = s1;
    f2[b * T_DIM + r] = s2;
    lmax = fmaxf(lmax, s2);
  }
  sred[threadIdx.x] = lmax;
  __syncthreads();
  for (int s = 128; s > 0; s >>= 1) {
    if (threadIdx.x < s)
      sred[threadIdx.x] = fmaxf(sred[threadIdx.x], sred[threadIdx.x + s]);
    __syncthreads();
  }
  if (threadIdx.x == 0) fmax[b] = sred[0];
}

// ---------------------------------------------------------------------------
// Kernel C: out = softmax_j(leaky(f1_i + f2_j)) @ h, fused single pass.
// Exact row max m_i = leaky(f1_i + max_j f2_j) (LeakyReLU is monotone).
// 1 WG = 1 batch x 128 rows (8 waves x 16 rows).
// h streamed through LDS in 64-row tiles, DOUBLE-BUFFERED via
// global_load_async_to_lds_b128 (ASYNCcnt), so tile t+1 DMA overlaps the
// WMMA loop on tile t.
// ---------------------------------------------------------------------------
__global__ __launch_bounds__(256) void gat_attn(const float* __restrict__ h,
                                                const float* __restrict__ f1,
                                                const float* __restrict__ f2,
                                                const float* __restrict__ fmax,
                                                float* __restrict__ out) {
  __shared__ __align__(16) float sh[2][TJ * LDSP];  // 2 x 33 KB ping-pong
  __shared__ float sf2[2][TJ];

  const int lane = threadIdx.x & 31;
  const int wave = threadIdx.x >> 5;
  const int half = lane >> 4;
  const int l16  = lane & 15;
  const int b    = blockIdx.y;
  const int row0 = blockIdx.x * 128 + wave * 16;

  const float f1v  = f1[b * T_DIM + row0 + l16];   // A-lane row = l16
  const float zmax = f1v + fmax[b];
  const float mrow = zmax >= 0.f ? zmax : ALPHA * zmax;  // exact row max of e

  // Per-thread slice for tile staging: 64x32 float4 chunks / 256 threads = 8.
  const int jr0 = threadIdx.x >> 5;       // row of my k=0 chunk
  const int c4  = threadIdx.x & 31;       // float4 column
  const float* hb = h + (size_t)b * T_DIM * OUT_F;

  auto issue_tile = [&](int j0, int buf) {
    uint32_t ldsb = (uint32_t)(uintptr_t)(&sh[buf][0]);
    #pragma unroll
    for (int k = 0; k < 8; ++k) {
      const int jr = jr0 + k * 8;
      async_ld_b128(ldsb + (uint32_t)(jr * LDSP + c4 * 4) * 4u,
                    hb + (size_t)(j0 + jr) * OUT_F + c4 * 4);
    }
    if (threadIdx.x < TJ) sf2[buf][threadIdx.x] = f2[b * T_DIM + j0 + threadIdx.x];
  };

  v8f acc[8] = {};
  float psum = 0.f;  // running softmax denominator contribution of this lane

  issue_tile(0, 0);

  for (int t = 0; t < T_DIM / TJ; ++t) {
    const int cur = t & 1;
    wait_async0();      // my DMA into sh[cur] complete
    __syncthreads();    // everyone's DMA done + prior compute on sh[cur^1] done
    if (t + 1 < T_DIM / TJ) issue_tile((t + 1) * TJ, cur ^ 1);

    const float* shc = &sh[cur][0];
    const float* sfc = &sf2[cur][0];

    #pragma unroll 4
    for (int jj = 0; jj < TJ; jj += 4) {
      const int jA = jj + 2 * half;  // K index held by this lane (A fragment)
      float z0 = f1v + sfc[jA];
      float z1 = f1v + sfc[jA + 1];
      float e0 = z0 >= 0.f ? z0 : ALPHA * z0;
      float e1 = z1 >= 0.f ? z1 : ALPHA * z1;
      v2f pa;
      pa.x = __expf(e0 - mrow);
      pa.y = __expf(e1 - mrow);
      psum += pa.x + pa.y;

      const float* b0 = shc + jA * LDSP;
      const float* b1 = b0 + LDSP;
      #pragma unroll
      for (int nt = 0; nt < 8; ++nt) {
        v2f pb;
        pb.x = b0[nt * 16 + l16];
        pb.y = b1[nt * 16 + l16];
        acc[nt] = wmma_f32(pa, pb, acc[nt]);
      }
    }
  }

  // Denominator: lanes L and L+16 both contributed to row L%16
  float d = psum + __shfl_xor(psum, 16, 32);  // lane L holds d_{L%16}
  float invd = 1.f / d;

  #pragma unroll
  for (int r = 0; r < 8; ++r) {
    float s = __shfl(invd, r + 8 * half, 32);  // d for output row M = r + 8*half
    float* orow = out + ((size_t)b * T_DIM + row0 + r + 8 * half) * OUT_F;
    #pragma unroll
    for (int nt = 0; nt < 8; ++nt)
      orow[nt * 16 + l16] = acc[nt][r] * s;
  }
}

// ---------------------------------------------------------------------------
extern "C" void kernel_launch(void* const* d_in, const int* in_sizes, int n_in,
                              void* d_out, int out_size, void* d_ws, size_t ws_size,
                              hipStream_t stream) {
  (void)in_sizes; (void)n_in; (void)out_size; (void)ws_size;
  const float* x = (const float*)d_in[0];
  const float* W = (const float*)d_in[1];
  const float* a = (const float*)d_in[2];
  float* out = (float*)d_out;

  // Workspace: h (8 MB) + f1 + f2 + fmax
  float* h  = (float*)d_ws;
  float* f1 = h  + (size_t)B_DIM * T_DIM * OUT_F;
  float* f2 = f1 + (size_t)B_DIM * T_DIM;
  float* fm = f2 + (size_t)B_DIM * T_DIM;

  gat_xw  <<<dim3(B_DIM * T_DIM / 128), 256, 0, stream>>>(x, W, h);
  gat_fvec<<<dim3(B_DIM),               256, 0, stream>>>(h, a, f1, f2, fm);
  gat_attn<<<dim3(T_DIM / 128, B_DIM),  256, 0, stream>>>(h, f1, f2, fm, out);
}